// Net_38079180047153
// MI455X (gfx1250) — compile-verified
//
#include <hip/hip_runtime.h>
#include <hip/hip_bf16.h>

typedef __attribute__((ext_vector_type(2))) float v2f;
typedef __attribute__((ext_vector_type(8))) float v8f;

#define N_NODES 100000
#define N_FEAT  128
#define HIDDEN  64
#define OUT_F   16

// ---------------------------------------------------------------------------
// Degree / normalization
// ---------------------------------------------------------------------------
__global__ __launch_bounds__(256) void deg_accum_kernel(const int* __restrict__ dst,
                                                        float* __restrict__ deg,
                                                        int nEdges) {
    int e = blockIdx.x * 256 + threadIdx.x;
    if (e < nEdges) atomicAdd(&deg[dst[e]], 1.0f);
}

__global__ __launch_bounds__(256) void dis_kernel(float* __restrict__ deg_to_dis, int n) {
    int i = blockIdx.x * 256 + threadIdx.x;
    if (i < n) deg_to_dis[i] = rsqrtf(deg_to_dis[i] + 1.0f);
}

// ---------------------------------------------------------------------------
// GEMM1: h1[N,64] = x[N,128] @ W1[128,64]  via V_WMMA_F32_16X16X4_F32
// 256 threads = 8 waves; each wave -> 16 rows x 64 cols (4 N-tiles).
// x is single-use (51 MB) -> non-temporal loads so it doesn't evict the
// reusable 25.6 MB h1/agg working set from the 192 MB L2.
// ---------------------------------------------------------------------------
__global__ __launch_bounds__(256) void gemm1_wmma_kernel(const float* __restrict__ x,
                                                         const float* __restrict__ W1,
                                                         float* __restrict__ h1) {
    __shared__ float lw[N_FEAT * HIDDEN];  // 32 KB
    for (int i = threadIdx.x; i < N_FEAT * HIDDEN; i += 256) lw[i] = W1[i];
    __syncthreads();

    const int wave = threadIdx.x >> 5;
    const int lane = threadIdx.x & 31;
    const int half = lane >> 4;   // 0: lanes 0-15, 1: lanes 16-31
    const int l16  = lane & 15;

    const int rowBase = blockIdx.x * 128 + wave * 16;

    v8f acc[4] = {};

    int arow = rowBase + l16;
    if (arow >= N_NODES) arow = N_NODES - 1;       // clamp tail loads (stores guarded)
    const float* xr = x + (long)arow * N_FEAT;

    for (int k = 0; k < N_FEAT; k += 4) {
        // A frag: VGPR0 = K=k (lo lanes) / K=k+2 (hi lanes); VGPR1 = K=k+1 / K=k+3
        v2f a;
        a.x = __builtin_nontemporal_load(xr + k + 2 * half + 0);
        a.y = __builtin_nontemporal_load(xr + k + 2 * half + 1);
#pragma unroll
        for (int t = 0; t < 4; ++t) {
            const int n = t * 16 + l16;
            v2f b;
            b.x = lw[(k + 2 * half + 0) * HIDDEN + n];
            b.y = lw[(k + 2 * half + 1) * HIDDEN + n];
            acc[t] = __builtin_amdgcn_wmma_f32_16x16x4_f32(
                false, a, false, b, (short)0, acc[t], false, false);
        }
    }

#pragma unroll
    for (int j = 0; j < 8; ++j) {
        const int m = rowBase + half * 8 + j;
        if (m < N_NODES) {
#pragma unroll
            for (int t = 0; t < 4; ++t)
                h1[(long)m * HIDDEN + t * 16 + l16] = acc[t][j];
        }
    }
}

// ---------------------------------------------------------------------------
// GEMM2: h2[N,16] = z1[N,64] @ W2[64,16]
// ---------------------------------------------------------------------------
__global__ __launch_bounds__(256) void gemm2_wmma_kernel(const float* __restrict__ z1,
                                                         const float* __restrict__ W2,
                                                         float* __restrict__ h2) {
    __shared__ float lw[HIDDEN * OUT_F];  // 4 KB
    for (int i = threadIdx.x; i < HIDDEN * OUT_F; i += 256) lw[i] = W2[i];
    __syncthreads();

    const int wave = threadIdx.x >> 5;
    const int lane = threadIdx.x & 31;
    const int half = lane >> 4;
    const int l16  = lane & 15;

    const int rowBase = blockIdx.x * 128 + wave * 16;

    v8f acc = {};

    int arow = rowBase + l16;
    if (arow >= N_NODES) arow = N_NODES - 1;
    const float* zr = z1 + (long)arow * HIDDEN;

    for (int k = 0; k < HIDDEN; k += 4) {
        v2f a;
        a.x = zr[k + 2 * half + 0];
        a.y = zr[k + 2 * half + 1];
        v2f b;
        b.x = lw[(k + 2 * half + 0) * OUT_F + l16];
        b.y = lw[(k + 2 * half + 1) * OUT_F + l16];
        acc = __builtin_amdgcn_wmma_f32_16x16x4_f32(
            false, a, false, b, (short)0, acc, false, false);
    }

#pragma unroll
    for (int j = 0; j < 8; ++j) {
        const int m = rowBase + half * 8 + j;
        if (m < N_NODES) h2[(long)m * OUT_F + l16] = acc[j];
    }
}

// ---------------------------------------------------------------------------
// Edge aggregation: agg[dst] += h[src] * dis[src]*dis[dst]
// FEAT/4 threads per edge, float4 gather + 4 float atomics.
// Only the group leader (g==0) loads src/dst/dis; (s,d,c) broadcast via
// sub-wave shuffles (lane-permute, no LDS traffic) -> cuts redundant VMEM
// ops in the dominant, bandwidth-bound kernel.
// ---------------------------------------------------------------------------
template <int FEAT>
__global__ __launch_bounds__(256) void aggregate_kernel(const float* __restrict__ h,
                                                        const int* __restrict__ src,
                                                        const int* __restrict__ dst,
                                                        const float* __restrict__ dis,
                                                        float* __restrict__ agg,
                                                        int nEdges) {
    constexpr int TPE = FEAT / 4;  // threads per edge (16 or 4; power of two)
    const long t = (long)blockIdx.x * 256 + threadIdx.x;
    const int e = (int)(t / TPE);
    const int g = (int)(t % TPE);
    if (e >= nEdges) return;       // whole TPE-group exits together (same e)

    int s = 0, d = 0;
    float c = 0.0f;
    if (g == 0) {
        s = src[e];
        d = dst[e];
        c = dis[s] * dis[d];
    }
    s = __shfl(s, 0, TPE);
    d = __shfl(d, 0, TPE);
    c = __shfl(c, 0, TPE);

    const float4 h4 = *(const float4*)(h + (long)s * FEAT + g * 4);
    float* o = agg + (long)d * FEAT + g * 4;
    atomicAdd(o + 0, h4.x * c);
    atomicAdd(o + 1, h4.y * c);
    atomicAdd(o + 2, h4.z * c);
    atomicAdd(o + 3, h4.w * c);
}

// ---------------------------------------------------------------------------
// Finalize: out = agg + h*dis^2 + b  (optional relu), in place over agg
// ---------------------------------------------------------------------------
template <int FEAT, bool RELU>
__global__ __launch_bounds__(256) void finalize_kernel(const float* __restrict__ h,
                                                       const float* __restrict__ dis,
                                                       const float* __restrict__ bias,
                                                       float* __restrict__ agg) {
    const long idx = (long)blockIdx.x * 256 + threadIdx.x;
    if (idx >= (long)N_NODES * FEAT) return;
    const int i = (int)(idx / FEAT);
    const int f = (int)(idx % FEAT);
    const float dd = dis[i];
    float v = agg[idx] + h[idx] * dd * dd + bias[f];
    if (RELU) v = v > 0.0f ? v : 0.0f;
    agg[idx] = v;
}

// ---------------------------------------------------------------------------
// Link prediction: logits[e] = dot(z[a], z[b]) over 16 dims.
// z (6.4 MB) stays hot in L2; the output is streamed with NT stores.
// ---------------------------------------------------------------------------
__global__ __launch_bounds__(256) void logits_kernel(const float* __restrict__ z,
                                                     const int* __restrict__ pos,
                                                     const int* __restrict__ neg,
                                                     float* __restrict__ out,
                                                     int P, int Q) {
    const int t = blockIdx.x * 256 + threadIdx.x;
    if (t >= P + Q) return;
    int a, b;
    if (t < P) { a = pos[t]; b = pos[P + t]; }
    else       { int u = t - P; a = neg[u]; b = neg[Q + u]; }
    const float4* za = (const float4*)(z + (long)a * OUT_F);
    const float4* zb = (const float4*)(z + (long)b * OUT_F);
    float s = 0.0f;
#pragma unroll
    for (int i = 0; i < 4; ++i) {
        const float4 xa = za[i];
        const float4 xb = zb[i];
        s += xa.x * xb.x + xa.y * xb.y + xa.z * xb.z + xa.w * xb.w;
    }
    __builtin_nontemporal_store(s, out + t);
}

// ---------------------------------------------------------------------------
// Host launcher
// ---------------------------------------------------------------------------
extern "C" void kernel_launch(void* const* d_in, const int* in_sizes, int n_in,
                              void* d_out, int out_size, void* d_ws, size_t ws_size,
                              hipStream_t stream) {
    const float* x   = (const float*)d_in[0];
    const int*   ei  = (const int*)d_in[1];
    const int*   pos = (const int*)d_in[2];
    const int*   neg = (const int*)d_in[3];
    const float* W1  = (const float*)d_in[4];
    const float* b1  = (const float*)d_in[5];
    const float* W2  = (const float*)d_in[6];
    const float* b2  = (const float*)d_in[7];
    float* out = (float*)d_out;

    const int E = in_sizes[1] / 2;
    const int P = in_sizes[2] / 2;
    const int Q = in_sizes[3] / 2;
    const int* src = ei;
    const int* dst = ei + E;

    // Workspace layout (bytes):
    //   [0 .. 400000)                    deg -> dis (in place)
    //   [524288 .. +25.6MB)              h1   (reused as h2 + agg2 after layer 1)
    //   [524288+25.6MB .. +25.6MB)       agg1 -> z1 (in place)
    char* ws = (char*)d_ws;
    const size_t OFF_H1   = 524288;
    const size_t SZ_H1    = (size_t)N_NODES * HIDDEN * sizeof(float);  // 25.6 MB
    const size_t OFF_AGG1 = OFF_H1 + SZ_H1;
    const size_t SZ_H2    = (size_t)N_NODES * OUT_F * sizeof(float);   // 6.4 MB

    float* dis  = (float*)ws;
    float* h1   = (float*)(ws + OFF_H1);
    float* agg1 = (float*)(ws + OFF_AGG1);       // becomes z1
    float* h2   = (float*)(ws + OFF_H1);         // reuse h1 slot
    float* agg2 = (float*)(ws + OFF_H1 + SZ_H2); // becomes z
    float* z1   = agg1;
    float* z    = agg2;

    auto cdiv = [](long a, long b) { return (int)((a + b - 1) / b); };

    // 1. degree -> dis
    hipMemsetAsync(dis, 0, (size_t)N_NODES * sizeof(float), stream);
    deg_accum_kernel<<<cdiv(E, 256), 256, 0, stream>>>(dst, dis, E);
    dis_kernel<<<cdiv(N_NODES, 256), 256, 0, stream>>>(dis, N_NODES);

    // 2. h1 = x @ W1  (WMMA, NT loads of x)
    gemm1_wmma_kernel<<<cdiv(N_NODES, 128), 256, 0, stream>>>(x, W1, h1);

    // 3. agg1 = scatter_add(h1[src] * coef, dst)
    hipMemsetAsync(agg1, 0, SZ_H1, stream);
    aggregate_kernel<HIDDEN><<<cdiv((long)E * (HIDDEN / 4), 256), 256, 0, stream>>>(
        h1, src, dst, dis, agg1, E);

    // 4. z1 = relu(agg1 + h1*dis^2 + b1)
    finalize_kernel<HIDDEN, true><<<cdiv((long)N_NODES * HIDDEN, 256), 256, 0, stream>>>(
        h1, dis, b1, agg1);

    // 5. h2 = z1 @ W2  (WMMA)
    gemm2_wmma_kernel<<<cdiv(N_NODES, 128), 256, 0, stream>>>(z1, W2, h2);

    // 6. agg2 = scatter_add(h2[src] * coef, dst)
    hipMemsetAsync(agg2, 0, SZ_H2, stream);
    aggregate_kernel<OUT_F><<<cdiv((long)E * (OUT_F / 4), 256), 256, 0, stream>>>(
        h2, src, dst, dis, agg2, E);

    // 7. z = agg2 + h2*dis^2 + b2
    finalize_kernel<OUT_F, false><<<cdiv((long)N_NODES * OUT_F, 256), 256, 0, stream>>>(
        h2, dis, b2, agg2);

    // 8. logits
    logits_kernel<<<cdiv(P + Q, 256), 256, 0, stream>>>(z, pos, neg, out, P, Q);
}